// SenderGRULM_86225763434779
// MI455X (gfx1250) — compile-verified
//
#include <hip/hip_runtime.h>
#include <math.h>

typedef float v2f __attribute__((ext_vector_type(2)));
typedef float v8f __attribute__((ext_vector_type(8)));

#define BLK     256
#define MT      16          // batch rows per workgroup
#define HDIM    512
#define H3      1536
#define K2R     1024
#define VF      1026
#define NT_OUT  65          // ceil(1026/16), padded N = 1040
#define AST     516         // h_lds row stride (stride % 64 == 4 -> conflict-free)
#define SST     1044        // s_lds row stride
#define PST     1028        // proto staging stride
#define TSTEPS  14
#define BATCH   4096

// D = A(16x4,f32) * B(4x16,f32) + C(16x16,f32)
__device__ __forceinline__ v8f wmma_f32(v2f a, v2f b, v8f c) {
  return __builtin_amdgcn_wmma_f32_16x16x4_f32(false, a, false, b, (short)0, c,
                                               false, false);
}

// ---- pack W[N,K] (row major) as B-matrix (B[k][n] = W[n][k]) WMMA fragments:
// dst[(nt*KT+kt)*64 + lane*2 + v] = B[kt*4 + 2*(lane>>4) + v][nt*16 + (lane&15)]
__global__ void pack_frags_kernel(const float* __restrict__ W,
                                  float* __restrict__ dst, int N, int K, int NT,
                                  int KT) {
  size_t total = (size_t)NT * KT * 64;
  size_t idx = (size_t)blockIdx.x * blockDim.x + threadIdx.x;
  if (idx >= total) return;
  int vv = (int)(idx & 1);
  int L = (int)((idx >> 1) & 31);
  size_t f = idx >> 6;
  int kt = (int)(f % (size_t)KT);
  int nt = (int)(f / (size_t)KT);
  int n = nt * 16 + (L & 15);
  int k = kt * 4 + 2 * (L >> 4) + vv;
  dst[idx] = (n < N) ? W[(size_t)n * K + k] : 0.0f;
}

// ---- G = E @ W_ih^T : [1026,256] x [256,1536] -> [1026,1536]
__global__ __launch_bounds__(256) void compute_G_kernel(
    const float* __restrict__ E, const float* __restrict__ Wf_ih,
    float* __restrict__ G) {
  const int w = threadIdx.x >> 5, lane = threadIdx.x & 31;
  const int mrow = lane & 15, half = lane >> 4;
  const int mt = blockIdx.x;            // 0..64
  const int nt = blockIdx.y * 8 + w;    // 0..95
  int mload = mt * 16 + mrow;
  if (mload > VF - 1) mload = VF - 1;   // clamp padded rows (results unused)
  const float* Erow = E + (size_t)mload * 256;
  v8f c = {0.f, 0.f, 0.f, 0.f, 0.f, 0.f, 0.f, 0.f};
  for (int k0 = 0; k0 < 256; k0 += 4) {
    v2f a = *(const v2f*)(Erow + k0 + 2 * half);
    v2f b = ((const v2f*)&Wf_ih[((size_t)nt * 64 + (k0 >> 2)) * 64])[lane];
    c = wmma_f32(a, b, c);
  }
#pragma unroll
  for (int r = 0; r < 8; ++r) {
    int m = mt * 16 + r + 8 * half;
    if (m < VF) G[(size_t)m * H3 + nt * 16 + mrow] = c[r];
  }
}

__global__ void zero_f4(float4* __restrict__ p, size_t n4) {
  size_t i = (size_t)blockIdx.x * blockDim.x + threadIdx.x;
  if (i < n4) p[i] = make_float4(0.f, 0.f, 0.f, 0.f);
}

// ---- persistent GRU: one WG owns 16 batch rows for all 14 steps ----
__global__ __launch_bounds__(BLK) void sender_gru_kernel(
    const float* __restrict__ proto0, const float* __restrict__ proto1,
    const float* __restrict__ gum, const float* __restrict__ b_init,
    const float* __restrict__ b_ih, const float* __restrict__ b_hh,
    const float* __restrict__ b_out, const float* __restrict__ Wf_init,
    const float* __restrict__ Wf_hh, const float* __restrict__ Wf_out,
    const float* __restrict__ G, float* __restrict__ out) {
  __shared__ float h_lds[MT * AST];   // 33 KB  : hidden state tile [16][512]
  __shared__ float s_lds[MT * SST];   // 66.8 KB: logits tile / proto staging
  __shared__ float stval[MT];
  __shared__ int sidx[MT];
  __shared__ float pmax[MT * 16];
  __shared__ float psum[MT * 16];
  __shared__ int pidx[MT * 16];

  const int tid = threadIdx.x;
  const int w = tid >> 5;
  const int lane = tid & 31;
  const int mrow = lane & 15;
  const int half = lane >> 4;
  const int b0 = blockIdx.x * MT;

  if (tid < MT) {
    stval[tid] = 1.0f;   // x0 = exact one-hot SOS
    sidx[tid] = 0;       // SOS_IDX
    size_t rb = (size_t)(b0 + tid) * 16 * VF;
    out[rb + 0] = 1.0f;                // SOS at t=0, v=0
    out[rb + (size_t)15 * VF + 1] = 1.0f;  // EOS at t=15, v=1
  }

  // stage concat(proto0,proto1) rows into s_lds (stride PST)
  for (int i = tid; i < MT * K2R; i += BLK) {
    int m = i >> 10;
    int c = i & 1023;
    float v = (c < 512) ? proto0[(size_t)(b0 + m) * 512 + c]
                        : proto1[(size_t)(b0 + m) * 512 + (c - 512)];
    s_lds[m * PST + c] = v;
  }
  __syncthreads();

  // h0 = concat @ W_init^T + b_init  (32 n-tiles, 4 per wave)
  for (int q = 0; q < 4; ++q) {
    int nt = w * 4 + q;
    float bv = b_init[nt * 16 + mrow];
    v8f c;
#pragma unroll
    for (int r = 0; r < 8; ++r) c[r] = bv;
    for (int k0 = 0; k0 < K2R; k0 += 4) {
      v2f a = *(const v2f*)&s_lds[mrow * PST + k0 + 2 * half];
      v2f b = ((const v2f*)&Wf_init[((size_t)nt * 256 + (k0 >> 2)) * 64])[lane];
      c = wmma_f32(a, b, c);
    }
#pragma unroll
    for (int r = 0; r < 8; ++r)
      h_lds[(r + 8 * half) * AST + nt * 16 + mrow] = c[r];
  }
  __syncthreads();

  for (int t = 0; t < TSTEPS; ++t) {
    // -- phase A: gh = h @ W_hh^T + b_hh for this wave's 64 hidden columns,
    //    fused gates; gh never materialized in LDS.
    float hnew[4][8];
    {
      v8f cr[4], cz[4], cn[4];
#pragma unroll
      for (int jj = 0; jj < 4; ++jj) {
        int colb = (w * 4 + jj) * 16 + mrow;
        float br = b_hh[colb], bz = b_hh[512 + colb], bn = b_hh[1024 + colb];
#pragma unroll
        for (int r = 0; r < 8; ++r) {
          cr[jj][r] = br; cz[jj][r] = bz; cn[jj][r] = bn;
        }
      }
      for (int k0 = 0; k0 < HDIM; k0 += 4) {
        v2f a = *(const v2f*)&h_lds[mrow * AST + k0 + 2 * half];
        int kt = k0 >> 2;
#pragma unroll
        for (int jj = 0; jj < 4; ++jj) {
          int nr = w * 4 + jj;
          v2f brf = ((const v2f*)&Wf_hh[((size_t)nr * 128 + kt) * 64])[lane];
          v2f bzf = ((const v2f*)&Wf_hh[((size_t)(32 + nr) * 128 + kt) * 64])[lane];
          v2f bnf = ((const v2f*)&Wf_hh[((size_t)(64 + nr) * 128 + kt) * 64])[lane];
          cr[jj] = wmma_f32(a, brf, cr[jj]);
          cz[jj] = wmma_f32(a, bzf, cz[jj]);
          cn[jj] = wmma_f32(a, bnf, cn[jj]);
        }
      }
#pragma unroll
      for (int jj = 0; jj < 4; ++jj) {
        int col = (w * 4 + jj) * 16 + mrow;  // hidden index 0..511
#pragma unroll
        for (int r = 0; r < 8; ++r) {
          int m = r + 8 * half;
          float sv = stval[m];
          const float* Grow = G + (size_t)sidx[m] * H3;
          float gir = fmaf(sv, Grow[col], b_ih[col]);
          float giz = fmaf(sv, Grow[col + 512], b_ih[col + 512]);
          float gin = fmaf(sv, Grow[col + 1024], b_ih[col + 1024]);
          float rg = 1.0f / (1.0f + expf(-(gir + cr[jj][r])));
          float zg = 1.0f / (1.0f + expf(-(giz + cz[jj][r])));
          float ng = tanhf(gin + rg * cn[jj][r]);
          float ho = h_lds[m * AST + col];
          hnew[jj][r] = (1.0f - zg) * ng + zg * ho;
        }
      }
    }
    __syncthreads();  // all waves done reading old h
#pragma unroll
    for (int jj = 0; jj < 4; ++jj) {
      int col = (w * 4 + jj) * 16 + mrow;
#pragma unroll
      for (int r = 0; r < 8; ++r)
        h_lds[(r + 8 * half) * AST + col] = hnew[jj][r];
    }
    __syncthreads();

    // -- phase B: logits = h_new @ W_out^T + b_out (N padded to 1040)
    for (int q = 0; q < 9; ++q) {
      int nt = w + 8 * q;
      if (nt >= NT_OUT) break;  // wave-uniform
      int ncol = nt * 16 + mrow;
      float bv = (ncol < VF) ? b_out[ncol] : 0.0f;
      v8f c;
#pragma unroll
      for (int r = 0; r < 8; ++r) c[r] = bv;
      for (int k0 = 0; k0 < HDIM; k0 += 4) {
        v2f a = *(const v2f*)&h_lds[mrow * AST + k0 + 2 * half];
        v2f b = ((const v2f*)&Wf_out[((size_t)nt * 128 + (k0 >> 2)) * 64])[lane];
        c = wmma_f32(a, b, c);
      }
#pragma unroll
      for (int r = 0; r < 8; ++r)
        s_lds[(r + 8 * half) * SST + ncol] = c[r];
    }
    __syncthreads();

    // -- phase C: gumbel noise + online argmax/softmax, scatter token
    {
      int m = tid >> 4;
      int lr = tid & 15;
      const float* urow = gum + ((size_t)t * BATCH + b0 + m) * VF;
      float lmax = -INFINITY, lsum = 0.0f;
      int li = 0;
      for (int i = 0; i < 65; ++i) {
        int v = lr + 16 * i;
        if (v < VF) {
          float g = -logf(-logf(urow[v] + 1e-10f) + 1e-10f);
          float s = s_lds[m * SST + v] + g;  // TAU == 1.0
          if (s > lmax) {
            lsum = lsum * expf(lmax - s) + 1.0f;
            lmax = s;
            li = v;
          } else {
            lsum += expf(s - lmax);
          }
        }
      }
      pmax[m * 16 + lr] = lmax;
      psum[m * 16 + lr] = lsum;
      pidx[m * 16 + lr] = li;
    }
    __syncthreads();
    if (tid < MT) {
      int m = tid;
      float Mx = -INFINITY;
      int I = 0;
#pragma unroll
      for (int j = 0; j < 16; ++j) {
        float pm = pmax[m * 16 + j];
        if (pm > Mx) { Mx = pm; I = pidx[m * 16 + j]; }
      }
      float S = 0.0f;
#pragma unroll
      for (int j = 0; j < 16; ++j) S += psum[m * 16 + j] * expf(pmax[m * 16 + j] - Mx);
      float y = 0.9f * (1.0f / S) + (0.1f / 1024.0f);  // UNIFORM_W / V
      float stv = (1.0f - y) + y;                      // straight-through value
      stval[m] = stv;
      sidx[m] = I;
      out[((size_t)(b0 + m) * 16 + (t + 1)) * VF + I] = stv;
    }
    __syncthreads();
  }
}

extern "C" void kernel_launch(void* const* d_in, const int* in_sizes, int n_in,
                              void* d_out, int out_size, void* d_ws,
                              size_t ws_size, hipStream_t stream) {
  const float* proto0 = (const float*)d_in[0];
  const float* proto1 = (const float*)d_in[1];
  const float* gum = (const float*)d_in[2];
  const float* W_init = (const float*)d_in[3];
  const float* b_init = (const float*)d_in[4];
  const float* W_ih = (const float*)d_in[5];
  const float* b_ih = (const float*)d_in[6];
  const float* W_hh = (const float*)d_in[7];
  const float* b_hh = (const float*)d_in[8];
  const float* W_out = (const float*)d_in[9];
  const float* b_out = (const float*)d_in[10];
  const float* E = (const float*)d_in[11];
  float* out = (float*)d_out;

  float* ws = (float*)d_ws;
  float* Wf_hh = ws;                                   //  96*128*64
  float* Wf_out = Wf_hh + (size_t)96 * 128 * 64;       //  65*128*64
  float* Wf_init = Wf_out + (size_t)65 * 128 * 64;     //  32*256*64
  float* Wf_ih = Wf_init + (size_t)32 * 256 * 64;      //  96*64*64
  float* G = Wf_ih + (size_t)96 * 64 * 64;             //  1026*1536

  struct PackArg { const float* W; float* dst; int N; int K; };
  const PackArg packs[4] = {{W_hh, Wf_hh, 1536, 512},
                            {W_out, Wf_out, 1026, 512},
                            {W_init, Wf_init, 512, 1024},
                            {W_ih, Wf_ih, 1536, 256}};
  for (int i = 0; i < 4; ++i) {
    int NT = (packs[i].N + 15) / 16, KT = packs[i].K / 4;
    size_t total = (size_t)NT * KT * 64;
    int blocks = (int)((total + 255) / 256);
    pack_frags_kernel<<<blocks, 256, 0, stream>>>(packs[i].W, packs[i].dst,
                                                  packs[i].N, packs[i].K, NT, KT);
  }

  compute_G_kernel<<<dim3(65, 12), 256, 0, stream>>>(E, Wf_ih, G);

  size_t n4 = (size_t)BATCH * 16 * VF / 4;
  zero_f4<<<(unsigned)((n4 + 255) / 256), 256, 0, stream>>>((float4*)out, n4);

  sender_gru_kernel<<<dim3(BATCH / MT), BLK, 0, stream>>>(
      proto0, proto1, gum, b_init, b_ih, b_hh, b_out, Wf_init, Wf_hh, Wf_out, G,
      out);
}